// GAT_38714835206732
// MI455X (gfx1250) — compile-verified
//
#include <hip/hip_runtime.h>
#include <hip/hip_bf16.h>

typedef float v2f __attribute__((ext_vector_type(2)));
typedef float v8f __attribute__((ext_vector_type(8)));

#define GAT_IN_DIM 128
#define GAT_NEG_SLOPE 0.2f

// ---------------------------------------------------------------------------
// f32 WMMA GEMM: C[M,N] = A[M,K] * B[K,N] (+ optional bias[N])
// One wave computes one 16x16 tile via V_WMMA_F32_16X16X4_F32, K stepped by 4.
// Layout (ISA 7.12.2, 32-bit): A/B element at vgpr j, lane l -> k = 2*(l>>4)+j,
// row/col = l&15. D vgpr r -> M = r + 8*(l>>4), N = l&15.
// ---------------------------------------------------------------------------
__global__ void wmma_gemm_f32(const float* __restrict__ A,
                              const float* __restrict__ B,
                              float* __restrict__ C,
                              const float* __restrict__ bias,
                              int M, int N, int K) {
  int wave = (blockIdx.x * blockDim.x + threadIdx.x) >> 5;
  int lane = threadIdx.x & 31;
  int tiles_n = N >> 4;
  int tiles = (M >> 4) * tiles_n;
  if (wave >= tiles) return;
  int tm = (wave / tiles_n) << 4;
  int tn = (wave % tiles_n) << 4;
  int idx  = lane & 15;        // M for A, N for B/D
  int kk   = (lane >> 4) << 1; // K pair base {0,1} or {2,3}
  int half = lane >> 4;

  v8f acc = {};
  for (int k0 = 0; k0 < K; k0 += 4) {
    v2f a, b;
    const float* ap = A + (size_t)(tm + idx) * K + (k0 + kk);
    a.x = ap[0];
    a.y = ap[1];
    const float* bp = B + (size_t)(k0 + kk) * N + tn + idx;
    b.x = bp[0];
    b.y = bp[N];
    acc = __builtin_amdgcn_wmma_f32_16x16x4_f32(
        /*neg_a=*/false, a, /*neg_b=*/false, b,
        /*c_mod=*/(short)0, acc, /*reuse_a=*/false, /*reuse_b=*/false);
  }

  float badd = bias ? bias[tn + idx] : 0.0f;
#pragma unroll
  for (int r = 0; r < 8; ++r) {
    C[(size_t)(tm + r + 8 * half) * N + tn + idx] = acc[r] + badd;
  }
}

// ---------------------------------------------------------------------------
// Per-(node, head) attention scores: es = <h, a_l>, ed = <h, a_r>
// attn layout: [H][2*F] with a_l = attn[h][0:F], a_r = attn[h][F:2F]
// ---------------------------------------------------------------------------
__global__ void gat_scores(const float* __restrict__ h,
                           const float* __restrict__ attn,
                           float* __restrict__ es, float* __restrict__ ed,
                           int n_nodes, int H, int F) {
  int t = blockIdx.x * blockDim.x + threadIdx.x;
  if (t >= n_nodes * H) return;
  int n = t / H;
  int hh = t - n * H;
  const float* hp = h + (size_t)n * H * F + (size_t)hh * F;
  const float* al = attn + (size_t)hh * 2 * F;
  const float* ar = al + F;
  float s = 0.f, d = 0.f;
  for (int f = 0; f < F; ++f) {
    s += hp[f] * al[f];
    d += hp[f] * ar[f];
  }
  es[t] = s;
  ed[t] = d;
}

// order-preserving float <-> uint for atomic segment-max
__device__ __forceinline__ unsigned encf(float f) {
  unsigned u = __float_as_uint(f);
  return (u & 0x80000000u) ? ~u : (u | 0x80000000u);
}
__device__ __forceinline__ float decf(unsigned u) {
  return (u & 0x80000000u) ? __uint_as_float(u & 0x7FFFFFFFu)
                           : __uint_as_float(~u);
}

// alpha = leaky_relu(es[row] + ed[col]); atomicMax over destination segments
__global__ void edge_alpha_kernel(const float* __restrict__ es,
                                  const float* __restrict__ ed,
                                  const int* __restrict__ rowE,
                                  const int* __restrict__ colE,
                                  float* __restrict__ alpha,
                                  unsigned* __restrict__ amax,
                                  int E, int Nn, int H) {
  int t = blockIdx.x * blockDim.x + threadIdx.x;
  if (t >= (E + Nn) * H) return;
  int e = t / H;
  int h = t - e * H;
  int r = (e < E) ? rowE[e] : (e - E);
  int c = (e < E) ? colE[e] : (e - E);
  float a = es[r * H + h] + ed[c * H + h];
  a = (a > 0.f) ? a : a * GAT_NEG_SLOPE;
  alpha[t] = a;
  atomicMax(&amax[c * H + h], encf(a));
}

// ex = exp(alpha - amax[col]); denom[col] += ex   (ex stored in-place)
__global__ void edge_exp_kernel(float* __restrict__ alpha,
                                const unsigned* __restrict__ amax,
                                float* __restrict__ denom,
                                const int* __restrict__ colE,
                                int E, int Nn, int H) {
  int t = blockIdx.x * blockDim.x + threadIdx.x;
  if (t >= (E + Nn) * H) return;
  int e = t / H;
  int h = t - e * H;
  int c = (e < E) ? colE[e] : (e - E);
  float am = decf(amax[c * H + h]);
  float ex = __expf(alpha[t] - am);
  alpha[t] = ex;
  atomicAdd(&denom[c * H + h], ex);
}

// out[col, f] += h[row, f] * (ex / denom[col]) ; thread per (edge, feature)
__global__ void edge_scatter_kernel(const float* __restrict__ ex,
                                    const float* __restrict__ denom,
                                    const float* __restrict__ hsrc,
                                    float* __restrict__ outd,
                                    const int* __restrict__ rowE,
                                    const int* __restrict__ colE,
                                    int E, int Nn, int H, int F) {
  int Ftot = H * F;
  long long t = (long long)blockIdx.x * blockDim.x + threadIdx.x;
  long long total = (long long)(E + Nn) * Ftot;
  if (t >= total) return;
  int e = (int)(t / Ftot);
  int f = (int)(t - (long long)e * Ftot);
  int h = f / F;
  int r = (e < E) ? rowE[e] : (e - E);
  int c = (e < E) ? colE[e] : (e - E);
  float coef = ex[(size_t)e * H + h] / denom[c * H + h];
  atomicAdd(&outd[(size_t)c * Ftot + f], hsrc[(size_t)r * Ftot + f] * coef);
}

__global__ void relu_kernel(float* __restrict__ p, long long n) {
  long long t = (long long)blockIdx.x * blockDim.x + threadIdx.x;
  if (t < n) p[t] = fmaxf(p[t], 0.f);
}

// ---------------------------------------------------------------------------

extern "C" void kernel_launch(void* const* d_in, const int* in_sizes, int n_in,
                              void* d_out, int out_size, void* d_ws, size_t ws_size,
                              hipStream_t stream) {
  const float* x     = (const float*)d_in[0];
  const int*   ei    = (const int*)  d_in[1];
  const float* W1    = (const float*)d_in[2];
  const float* attn1 = (const float*)d_in[3];
  const float* W2    = (const float*)d_in[4];
  const float* attn2 = (const float*)d_in[5];
  const float* headW = (const float*)d_in[6];
  const float* headb = (const float*)d_in[7];

  const int Nn = in_sizes[0] / GAT_IN_DIM; // 50000
  const int E  = in_sizes[1] / 2;          // 800000
  const int ET = E + Nn;                   // + self loops
  const int* rowE = ei;
  const int* colE = ei + E;

  // ---- workspace layout (floats) ----
  float* w = (float*)d_ws;
  float* h1    = w; w += (size_t)Nn * 128;
  float* out1  = w; w += (size_t)Nn * 128;
  float* h2    = w; w += (size_t)Nn * 64;
  float* out2  = w; w += (size_t)Nn * 64;
  float* es1   = w; w += (size_t)Nn * 4;
  float* ed1   = w; w += (size_t)Nn * 4;
  unsigned* amax1 = (unsigned*)w; w += (size_t)Nn * 4;
  float* den1  = w; w += (size_t)Nn * 4;
  float* es2   = w; w += (size_t)Nn;
  float* ed2   = w; w += (size_t)Nn;
  unsigned* amax2 = (unsigned*)w; w += (size_t)Nn;
  float* den2  = w; w += (size_t)Nn;
  float* ebuf  = w; w += (size_t)ET * 4;   // reused for layer 2 (H=1)

  hipMemsetAsync(out1,  0, (size_t)Nn * 128 * sizeof(float), stream);
  hipMemsetAsync(out2,  0, (size_t)Nn * 64  * sizeof(float), stream);
  hipMemsetAsync(amax1, 0, (size_t)Nn * 4   * sizeof(unsigned), stream);
  hipMemsetAsync(den1,  0, (size_t)Nn * 4   * sizeof(float), stream);
  hipMemsetAsync(amax2, 0, (size_t)Nn       * sizeof(unsigned), stream);
  hipMemsetAsync(den2,  0, (size_t)Nn       * sizeof(float), stream);

  const int TB = 256;                 // 8 waves/block
  auto gemm_blocks = [](int M, int N) {
    int tiles = (M >> 4) * (N >> 4);
    return (tiles + 7) / 8;
  };

  // ===== Layer 1: heads=4, f_out=32 =====
  wmma_gemm_f32<<<gemm_blocks(Nn, 128), TB, 0, stream>>>(x, W1, h1, nullptr, Nn, 128, 128);
  gat_scores<<<(Nn * 4 + TB - 1) / TB, TB, 0, stream>>>(h1, attn1, es1, ed1, Nn, 4, 32);
  {
    int nt = ET * 4;
    edge_alpha_kernel<<<(nt + TB - 1) / TB, TB, 0, stream>>>(es1, ed1, rowE, colE, ebuf, amax1, E, Nn, 4);
    edge_exp_kernel<<<(nt + TB - 1) / TB, TB, 0, stream>>>(ebuf, amax1, den1, colE, E, Nn, 4);
    long long ns = (long long)ET * 128;
    edge_scatter_kernel<<<(unsigned)((ns + TB - 1) / TB), TB, 0, stream>>>(
        ebuf, den1, h1, out1, rowE, colE, E, Nn, 4, 32);
  }
  {
    long long n = (long long)Nn * 128;
    relu_kernel<<<(unsigned)((n + TB - 1) / TB), TB, 0, stream>>>(out1, n);
  }

  // ===== Layer 2: heads=1, f_out=64 =====
  wmma_gemm_f32<<<gemm_blocks(Nn, 64), TB, 0, stream>>>(out1, W2, h2, nullptr, Nn, 64, 128);
  gat_scores<<<(Nn + TB - 1) / TB, TB, 0, stream>>>(h2, attn2, es2, ed2, Nn, 1, 64);
  {
    int nt = ET;
    edge_alpha_kernel<<<(nt + TB - 1) / TB, TB, 0, stream>>>(es2, ed2, rowE, colE, ebuf, amax2, E, Nn, 1);
    edge_exp_kernel<<<(nt + TB - 1) / TB, TB, 0, stream>>>(ebuf, amax2, den2, colE, E, Nn, 1);
    long long ns = (long long)ET * 64;
    edge_scatter_kernel<<<(unsigned)((ns + TB - 1) / TB), TB, 0, stream>>>(
        ebuf, den2, h2, out2, rowE, colE, E, Nn, 1, 64);
  }

  // ===== Head: out = out2 @ head_W + head_b =====
  wmma_gemm_f32<<<gemm_blocks(Nn, 64), TB, 0, stream>>>(out2, headW, (float*)d_out, headb, Nn, 64, 64);
}